// Voxelization_23845658427974
// MI455X (gfx1250) — compile-verified
//
#include <hip/hip_runtime.h>
#include <hip/hip_bf16.h>

// ---------------- problem constants (match reference) ----------------
#define GX 1408
#define GY 1600
#define GZ 40
#define MAXV 60000
#define MAXP 35

// hash table: 4M slots (load factor ~0.45 for ~1.9M valid points)
#define HBITS 22
#define HSIZE (1u << HBITS)
#define HMASK (HSIZE - 1u)

// points staged per block through LDS by the Tensor Data Mover
#define TILE_PTS 1024                    // 16 KB tile, 4 points / thread

// d_out layout (floats), concatenated in reference return order
#define COORS_OFF (MAXV * MAXP * 4)      // 8,400,000
#define NUMPV_OFF (COORS_OFF + MAXV * 3) // 8,580,000
#define VN_OFF    (NUMPV_OFF + MAXV)     // 8,640,000

typedef unsigned int u32x4 __attribute__((ext_vector_type(4)));
typedef int          i32x4 __attribute__((ext_vector_type(4)));
typedef int          i32x8 __attribute__((ext_vector_type(8)));

// ---------------- kernels ----------------

__global__ void vz_zero_out(float4* __restrict__ out4, int n4,
                            float* __restrict__ out, int out_size) {
  int i = blockIdx.x * blockDim.x + threadIdx.x;
  if (i < n4) out4[i] = make_float4(0.f, 0.f, 0.f, 0.f);
  if (i == 0) {
    for (int j = n4 * 4; j < out_size; ++j) out[j] = 0.f;  // <=3 tail floats
  }
}

__global__ void vz_init_ws(int* __restrict__ key, int* __restrict__ mini,
                           int* __restrict__ cnt, int* __restrict__ head,
                           int* __restrict__ S, int n) {
  unsigned i = blockIdx.x * blockDim.x + threadIdx.x;
  unsigned stride = gridDim.x * blockDim.x;
  for (unsigned j = i; j < HSIZE; j += stride) {
    key[j] = -1;
    mini[j] = 0x7fffffff;
    cnt[j] = 0;
    head[j] = -1;
  }
  for (unsigned j = i; j < (unsigned)n; j += stride) S[j] = 0;
}

// Phase 1: TDM stages a 16KB point tile into LDS (one tensor_load_to_lds per
// block, issued by wave 0, completion via TENSORcnt), then each thread
// hash-inserts 4 points. All per-slot reductions are commutative atomics
// (min/add) or order-free (linked-list build) -> deterministic final output.
__global__ __launch_bounds__(256) void vz_bin_points(
    const uint4* __restrict__ pts, int n,
    int* __restrict__ key, int* __restrict__ mini, int* __restrict__ cnt,
    int* __restrict__ head, int* __restrict__ next) {
  __shared__ uint4 sbuf[TILE_PTS];       // only LDS object -> offset 0
  const int tile0 = blockIdx.x * TILE_PTS;

  if (threadIdx.x < 32) {                // wave 0 issues the tensor DMA
    unsigned long long ga = (unsigned long long)(const void*)(pts + tile0);
    long long remll = ((long long)n - (long long)tile0) * 4;   // dwords left
    unsigned td0 = (remll > 0) ? (unsigned)remll : 0u;         // tensor_dim0

    // ---- D# group 0: count=1 | lds_addr | global_addr[56:0] | type=2 ----
    u32x4 g0;
    g0.x = 1u;                                         // count=1 (user desc)
    g0.y = 0u;                                         // lds_addr = sbuf @ 0
    g0.z = (unsigned)(ga & 0xffffffffull);             // global_addr[31:0]
    g0.w = (unsigned)((ga >> 32) & 0x1ffffffull)       // global_addr[56:32]
         | (2u << 30);                                 // type = 2 ("image")
    // ---- D# group 1: 1-D tile, 4-byte elements ----
    // word0: wg_mask=0 | data_size=2 (4B)
    // tensor_dim0 = td0 (hardware zero-fills OOB reads -> free tail handling)
    // tile_dim0 = 4096 dwords (16KB); tile_dim1 = 0 (1-D)
    i32x8 g1;
    g1.s0 = (int)(2u << 16);
    g1.s1 = (int)((td0 & 0xffffu) << 16);              // tensor_dim0[15:0]
    g1.s2 = (int)((td0 >> 16) | (1u << 16));           // tensor_dim0[31:16], tensor_dim1=1
    g1.s3 = (int)(4096u << 16);                        // tile_dim0 = 4096
    g1.s4 = 0;                                         // tile_dim1/2 = 0
    g1.s5 = (int)td0;                                  // tensor_dim0_stride lo
    g1.s6 = 0;
    g1.s7 = 0;
    i32x4 z4 = {0, 0, 0, 0};
#if defined(__clang_major__) && (__clang_major__ >= 23)
    i32x8 z8 = {0, 0, 0, 0, 0, 0, 0, 0};
    __builtin_amdgcn_tensor_load_to_lds(g0, g1, z4, z4, z8, 0);
#else
    __builtin_amdgcn_tensor_load_to_lds(g0, g1, z4, z4, 0);
#endif
    __builtin_amdgcn_s_wait_tensorcnt(0);
  }
  __syncthreads();                       // tile visible to all 8 waves

  // cover the next tile's cachelines (global_prefetch_b8)
  __builtin_prefetch((const char*)(pts + tile0 + TILE_PTS) + threadIdx.x * 64,
                     0, 1);

#pragma unroll
  for (int k = 0; k < TILE_PTS / 256; ++k) {
    const int li = threadIdx.x + k * 256;
    const int i = tile0 + li;
    if (i >= n) break;
    uint4 raw = sbuf[li];

    float px = __uint_as_float(raw.x);
    float py = __uint_as_float(raw.y);
    float pz = __uint_as_float(raw.z);

    // c = floor((p - lo) / voxel), same arithmetic as the reference
    int cx = (int)floorf((px - 0.0f)  / 0.05f);
    int cy = (int)floorf((py + 40.0f) / 0.05f);
    int cz = (int)floorf((pz + 3.0f)  / 0.1f);
    bool ok = (cx >= 0) & (cx < GX) & (cy >= 0) & (cy < GY) &
              (cz >= 0) & (cz < GZ);
    if (!ok) continue;

    int lin = (cz * GY + cy) * GX + cx;

    // linear-probe hash insert (L2-resident table: 80MB < 192MB L2)
    unsigned h = ((unsigned)lin * 2654435761u) & HMASK;
    for (;;) {
      int kk = atomicCAS(&key[h], -1, lin);
      if (kk == -1 || kk == lin) break;
      h = (h + 1u) & HMASK;
    }
    atomicMin(&mini[h], i);              // group's first-appearance index
    atomicAdd(&cnt[h], 1);               // group's total valid-point count
    next[i] = atomicExch(&head[h], i);   // unordered member list
  }
}

// Phase 2: one mark per group at its first-appearance index.
// (Distinct groups have distinct first indices -> no races.)
__global__ void vz_mark(const int* __restrict__ key, const int* __restrict__ mini,
                        int* __restrict__ S) {
  unsigned s = blockIdx.x * blockDim.x + threadIdx.x;
  if (s >= HSIZE) return;
  if (key[s] != -1) S[mini[s]] = 1;
}

// Phase 3a: per-block exclusive scan of S (2048 elements / block, in place)
__global__ __launch_bounds__(256) void vz_scan1(int* __restrict__ S,
                                                int* __restrict__ bsums, int n) {
  __shared__ int sh[256];
  const int tid = threadIdx.x;
  const int base = blockIdx.x * 2048 + tid * 8;
  int v[8];
  int acc = 0;
#pragma unroll
  for (int j = 0; j < 8; ++j) {
    int idx = base + j;
    int x = (idx < n) ? S[idx] : 0;
    v[j] = acc;                          // thread-local exclusive prefix
    acc += x;
  }
  sh[tid] = acc;
  __syncthreads();
  for (int off = 1; off < 256; off <<= 1) {  // Hillis–Steele inclusive scan
    int t = (tid >= off) ? sh[tid - off] : 0;
    __syncthreads();
    sh[tid] += t;
    __syncthreads();
  }
  int excl = sh[tid] - acc;
  if (tid == 255) bsums[blockIdx.x] = sh[255];
#pragma unroll
  for (int j = 0; j < 8; ++j) {
    int idx = base + j;
    if (idx < n) S[idx] = excl + v[j];
  }
}

// Phase 3b: scan the (<=2048) block sums in one workgroup; emit group total
__global__ __launch_bounds__(1024) void vz_scan2(int* __restrict__ bsums, int nb,
                                                 int* __restrict__ total) {
  __shared__ int sh[1024];
  const int tid = threadIdx.x;
  const int i0 = tid * 2, i1 = tid * 2 + 1;
  int a = (i0 < nb) ? bsums[i0] : 0;
  int b = (i1 < nb) ? bsums[i1] : 0;
  int acc = a + b;
  sh[tid] = acc;
  __syncthreads();
  for (int off = 1; off < 1024; off <<= 1) {
    int t = (tid >= off) ? sh[tid - off] : 0;
    __syncthreads();
    sh[tid] += t;
    __syncthreads();
  }
  int excl = sh[tid] - acc;
  if (i0 < nb) bsums[i0] = excl;
  if (i1 < nb) bsums[i1] = excl + a;
  if (tid == 1023) *total = sh[1023];    // total number of valid groups
}

// Phase 4: one thread per occupied slot. vid = rank of first-appearance
// index; keep vid < MAXV. Walk the member list and select the MAXP smallest
// point indices in ascending order (deterministic), then gather/store.
__global__ __launch_bounds__(256) void vz_finalize(
    const float4* __restrict__ pts,
    const int* __restrict__ key, const int* __restrict__ mini,
    const int* __restrict__ cnt, const int* __restrict__ head,
    const int* __restrict__ next, const int* __restrict__ S,
    const int* __restrict__ bsums, float* __restrict__ out) {
  unsigned s = blockIdx.x * blockDim.x + threadIdx.x;
  if (s >= HSIZE) return;
  int lin = key[s];
  if (lin == -1) return;

  int m = mini[s];
  int vid = S[m] + bsums[m >> 11];       // global exclusive-scan value at m
  if (vid >= MAXV) return;               // overflow voxels dropped

  int c = cnt[s];
  int npv = (c < MAXP) ? c : MAXP;
  out[NUMPV_OFF + vid] = (float)npv;

  int z = lin / (GX * GY);
  int rem = lin - z * (GX * GY);
  int y = rem / GX;
  int x = rem - y * GX;
  out[COORS_OFF + vid * 3 + 0] = (float)z;  // stored (z, y, x)
  out[COORS_OFF + vid * 3 + 1] = (float)y;
  out[COORS_OFF + vid * 3 + 2] = (float)x;

  // select the MAXP smallest member indices, sorted ascending
  int idxs[MAXP];
  int k = 0;
  for (int cur = head[s]; cur != -1; cur = next[cur]) {
    if (k < MAXP) {
      int j = k++;
      while (j > 0 && idxs[j - 1] > cur) { idxs[j] = idxs[j - 1]; --j; }
      idxs[j] = cur;
    } else if (cur < idxs[MAXP - 1]) {
      int j = MAXP - 1;
      while (j > 0 && idxs[j - 1] > cur) { idxs[j] = idxs[j - 1]; --j; }
      idxs[j] = cur;
    }
  }

  float* dst = out + (size_t)vid * (MAXP * 4);
  for (int j = 0; j < k; ++j) {
    float4 p = pts[idxs[j]];
    dst[j * 4 + 0] = p.x;
    dst[j * 4 + 1] = p.y;
    dst[j * 4 + 2] = p.z;
    dst[j * 4 + 3] = p.w;
  }
}

__global__ void vz_write_vn(const int* __restrict__ total, float* __restrict__ out) {
  if (blockIdx.x == 0 && threadIdx.x == 0) {
    int t = *total;
    if (t > MAXV) t = MAXV;
    out[VN_OFF] = (float)t;
  }
}

// ---------------- launcher ----------------

extern "C" void kernel_launch(void* const* d_in, const int* in_sizes, int n_in,
                              void* d_out, int out_size, void* d_ws, size_t ws_size,
                              hipStream_t stream) {
  (void)n_in; (void)ws_size;
  const uint4*  pts_u = (const uint4*)d_in[0];
  const float4* pts_f = (const float4*)d_in[0];
  const int n = in_sizes[0] / 4;         // points are [N, 4] f32
  float* out = (float*)d_out;

  // workspace layout (ints): 4 hash arrays of HSIZE, next[N], S[N], bsums, total
  int* ws    = (int*)d_ws;
  int* key   = ws;
  int* mini  = ws + (size_t)HSIZE;
  int* cnt   = ws + (size_t)2 * HSIZE;
  int* head  = ws + (size_t)3 * HSIZE;
  int* nxt   = ws + (size_t)4 * HSIZE;
  int* S     = nxt + n;
  int* bsums = S + n;
  int* total = bsums + 2048;

  const int n4 = out_size >> 2;
  vz_zero_out<<<(n4 + 255) / 256, 256, 0, stream>>>((float4*)out, n4, out, out_size);
  vz_init_ws<<<2048, 256, 0, stream>>>(key, mini, cnt, head, S, n);

  const int nbbin = (n + TILE_PTS - 1) / TILE_PTS;
  vz_bin_points<<<nbbin, 256, 0, stream>>>(pts_u, n, key, mini, cnt, head, nxt);
  vz_mark<<<HSIZE / 256, 256, 0, stream>>>(key, mini, S);

  const int nb1 = (n + 2047) / 2048;     // <= 2048 for N <= 4.19M
  vz_scan1<<<nb1, 256, 0, stream>>>(S, bsums, n);
  vz_scan2<<<1, 1024, 0, stream>>>(bsums, nb1, total);

  vz_finalize<<<HSIZE / 256, 256, 0, stream>>>(pts_f, key, mini, cnt, head, nxt, S, bsums, out);
  vz_write_vn<<<1, 1, 0, stream>>>(total, out);
}